// TopKSparsitySTE_34248069219175
// MI455X (gfx1250) — compile-verified
//
#include <hip/hip_runtime.h>
#include <hip/hip_bf16.h>
#include <stdint.h>

// ---------------------------------------------------------------------------
// TopK (k = d/2) magnitude sparsification + L2 normalize, per row of 4096 f32.
// One 256-thread (8-wave, wave32) workgroup per row.
//   1) Stage row (16KB) into LDS via GLOBAL_LOAD_ASYNC_TO_LDS_B128 (ASYNCcnt).
//   2) Each thread owns 16 values in registers; keys = bits(|x|) (monotone).
//   3) Exact 4-pass radix select (8 bits/pass) for the 2048-th largest key:
//        per-wave privatized 256-bin LDS histograms (ds_add_u32),
//        suffix-scan over bins to locate digit, refine prefix/rank.
//   4) mask = key >= T ; scale = 1/(sqrt(sum(y^2)) + 1e-6) ; coalesced stores.
// ---------------------------------------------------------------------------

#define THREADS 256
#define ROW     4096
#define VPT     16          // values per thread
#define KSEL    2048        // k = 0.5 * 4096
#define NWAVES  8           // 256 / 32 (wave32)

#define AS1 __attribute__((address_space(1)))
#define AS3 __attribute__((address_space(3)))

typedef int v4i_t __attribute__((ext_vector_type(4)));

#if __has_builtin(__builtin_amdgcn_global_load_async_to_lds_b128)
#define HAVE_ASYNC_LDS 1
#endif

#if __has_builtin(__builtin_amdgcn_s_wait_asynccnt)
#define WAIT_ASYNC0() __builtin_amdgcn_s_wait_asynccnt(0)
#else
#define WAIT_ASYNC0() asm volatile("s_wait_asynccnt 0x0" ::: "memory")
#endif

__global__ __launch_bounds__(THREADS) void
topk_sparse_norm_kernel(const float* __restrict__ x, float* __restrict__ out)
{
    __shared__ float    s_row[ROW];                 // 16 KB staged row
    __shared__ uint32_t s_hist[NWAVES][256];        // 8 KB privatized histograms
    __shared__ uint32_t s_scan[2][256];             // 2 KB double-buffered scan
    __shared__ float    s_ws[NWAVES];
    __shared__ uint32_t s_bin, s_rem;
    __shared__ float    s_scale;

    const int  tid  = threadIdx.x;
    const int  wave = tid >> 5;                     // wave32: 8 waves / block
    const long row  = blockIdx.x;

    const float* __restrict__ xrow = x   + row * (long)ROW;
    float*       __restrict__ orow = out + row * (long)ROW;

    // ---- Stage the row into LDS ------------------------------------------
#if defined(HAVE_ASYNC_LDS)
    #pragma unroll
    for (int j = 0; j < 4; ++j) {
        const int chunk = tid + j * THREADS;        // float4 chunk id, 0..1023
        __builtin_amdgcn_global_load_async_to_lds_b128(
            (AS1 v4i_t*)(xrow + chunk * 4),
            (AS3 v4i_t*)(&s_row[chunk * 4]),
            /*offset=*/0, /*cpol=*/0);
    }
    WAIT_ASYNC0();
#else
    #pragma unroll
    for (int j = 0; j < 4; ++j) {
        const int chunk = tid + j * THREADS;
        *(float4*)(&s_row[chunk * 4]) = *(const float4*)(xrow + chunk * 4);
    }
#endif
    __syncthreads();

    // ---- Pull this thread's 16 values into registers ---------------------
    float    v[VPT];
    uint32_t key[VPT];
    #pragma unroll
    for (int j = 0; j < 4; ++j) {
        const float4 t4 = *(const float4*)(&s_row[(tid + j * THREADS) * 4]);
        v[4*j+0] = t4.x; v[4*j+1] = t4.y; v[4*j+2] = t4.z; v[4*j+3] = t4.w;
    }
    #pragma unroll
    for (int i = 0; i < VPT; ++i)
        key[i] = __float_as_uint(v[i]) & 0x7fffffffu;   // monotone key of |x|

    // ---- 4-pass radix select: exact bits of the KSEL-th largest key ------
    uint32_t prefix = 0;
    uint32_t rem    = KSEL;     // 1-based rank of target within remaining set
    const int r = 255 - tid;    // reversed index for suffix scan

    #pragma unroll 1
    for (int pass = 0; pass < 4; ++pass) {
        const int      shift  = 24 - pass * 8;
        const uint32_t himask = (pass == 0) ? 0u : (0xFFFFFFFFu << (shift + 8));

        #pragma unroll
        for (int w = 0; w < NWAVES; ++w) s_hist[w][tid] = 0;
        __syncthreads();

        #pragma unroll
        for (int i = 0; i < VPT; ++i) {
            if ((key[i] & himask) == prefix)    // still a candidate
                atomicAdd(&s_hist[wave][(key[i] >> shift) & 0xFFu], 1u);
        }
        __syncthreads();

        uint32_t c = 0;
        #pragma unroll
        for (int w = 0; w < NWAVES; ++w) c += s_hist[w][tid];

        // inclusive suffix sums over bins: S[t] = sum_{b >= t} counts[b]
        s_scan[0][r] = c;
        __syncthreads();
        int src = 0;
        #pragma unroll
        for (int off = 1; off < 256; off <<= 1) {
            uint32_t val = s_scan[src][r];
            if (r >= off) val += s_scan[src][r - off];
            s_scan[src ^ 1][r] = val;
            src ^= 1;
            __syncthreads();
        }
        const uint32_t cge = s_scan[src][255 - tid];                     // digit >= tid
        const uint32_t cgt = (tid == 255) ? 0u : s_scan[src][254 - tid]; // digit >  tid

        if (cge >= rem && cgt < rem) {      // exactly one thread matches
            s_bin = (uint32_t)tid;
            s_rem = rem - cgt;
        }
        __syncthreads();
        prefix |= (s_bin << shift);
        rem     = s_rem;
        __syncthreads();
    }
    const uint32_t T = prefix;              // bits of k-th largest |x|

    // ---- sum of squares of kept elements ---------------------------------
    float ss = 0.0f;
    #pragma unroll
    for (int i = 0; i < VPT; ++i)
        if (key[i] >= T) ss += v[i] * v[i];

    #pragma unroll
    for (int off = 16; off > 0; off >>= 1)
        ss += __shfl_down(ss, off, 32);     // wave32 reduction
    if ((tid & 31) == 0) s_ws[wave] = ss;
    __syncthreads();
    if (tid == 0) {
        float tot = 0.0f;
        #pragma unroll
        for (int w = 0; w < NWAVES; ++w) tot += s_ws[w];
        s_scale = 1.0f / (sqrtf(tot) + 1e-6f);
    }
    __syncthreads();
    const float scale = s_scale;

    // ---- masked, scaled, coalesced b128 stores ---------------------------
    #pragma unroll
    for (int j = 0; j < 4; ++j) {
        float4 o;
        o.x = (key[4*j+0] >= T) ? v[4*j+0] * scale : 0.0f;
        o.y = (key[4*j+1] >= T) ? v[4*j+1] * scale : 0.0f;
        o.z = (key[4*j+2] >= T) ? v[4*j+2] * scale : 0.0f;
        o.w = (key[4*j+3] >= T) ? v[4*j+3] * scale : 0.0f;
        *(float4*)(orow + (tid + j * THREADS) * 4) = o;
    }
}

extern "C" void kernel_launch(void* const* d_in, const int* in_sizes, int n_in,
                              void* d_out, int out_size, void* d_ws, size_t ws_size,
                              hipStream_t stream)
{
    (void)n_in; (void)d_ws; (void)ws_size; (void)out_size;
    const float* x   = (const float*)d_in[0];
    float*       out = (float*)d_out;
    const int rows = in_sizes[0] / ROW;     // 4 * 4096 = 16384 rows
    topk_sparse_norm_kernel<<<rows, THREADS, 0, stream>>>(x, out);
}